// MemoryModule_81570018885819
// MI455X (gfx1250) — compile-verified
//
#include <hip/hip_runtime.h>
#include <math.h>

typedef float v2f __attribute__((ext_vector_type(2)));
typedef float v4f __attribute__((ext_vector_type(4)));
typedef float v8f __attribute__((ext_vector_type(8)));

#define NUM_NODES 500000
#define MEM_DIM   128
#define IN_DIM    256
#define G3        384        // 3 * MEM_DIM
#define TILE_M    32         // batch rows per block
#define XP        260        // x-tile LDS pitch (260 % 64 == 4 -> conflict-free, 8B aligned)
#define HP        132        // h-tile LDS pitch
#define GP        388        // gate-buffer LDS pitch

// Low 32 bits of a generic pointer to __shared__ = LDS byte offset
// (CDNA5 aperture rule).
__device__ __forceinline__ unsigned lds_off(const void* p) {
    return (unsigned)(uintptr_t)p;
}

// Async DMA: 16 bytes per lane, global -> LDS, tracked by ASYNCcnt.
__device__ __forceinline__ void async_load_b128(unsigned lds_byte_off, const void* gaddr) {
    asm volatile("global_load_async_to_lds_b128 %0, %1, off"
                 :: "v"(lds_byte_off), "v"(gaddr)
                 : "memory");
}
__device__ __forceinline__ void wait_asynccnt0() {
    asm volatile("s_wait_asynccnt 0x0" ::: "memory");
}

// ---------------------------------------------------------------------------
// Kernel 1: bulk copy memory -> out_memory, last_update -> out_last (float4)
// ---------------------------------------------------------------------------
__global__ __launch_bounds__(256) void tgn_copy_kernel(
    const float* __restrict__ memory, const float* __restrict__ last_update,
    float* __restrict__ out_mem, float* __restrict__ out_last,
    long n_mem4, long n_last4)
{
    long i = (long)blockIdx.x * blockDim.x + threadIdx.x;
    if (i < n_mem4) {
        ((v4f*)out_mem)[i] = ((const v4f*)memory)[i];
    } else {
        long j = i - n_mem4;
        if (j < n_last4) ((v4f*)out_last)[j] = ((const v4f*)last_update)[j];
    }
}

// ---------------------------------------------------------------------------
// Kernel 2: gather -> GRU (f32 WMMA 16x16x4) -> scatter
// ---------------------------------------------------------------------------
__global__ __launch_bounds__(256) void tgn_gru_kernel(
    const int*   __restrict__ node_ids,
    const float* __restrict__ messages,
    const float* __restrict__ timestamps,
    const float* __restrict__ memory,
    const float* __restrict__ W_ih,   // [384, 256] row-major
    const float* __restrict__ W_hh,   // [384, 128] row-major
    const float* __restrict__ b_ih,   // [384]
    const float* __restrict__ b_hh,   // [384]
    float* __restrict__ out_mem,
    float* __restrict__ out_last)
{
    __shared__ __align__(16) float xs [TILE_M * XP];   // messages tile 32x256
    __shared__ __align__(16) float hs [TILE_M * HP];   // gathered h tile 32x128
    __shared__ __align__(16) float gis[TILE_M * GP];   // gi = x@W_ih^T + b_ih, 32x384
    __shared__ __align__(16) float ghs[TILE_M * GP];   // gh = h@W_hh^T + b_hh, 32x384
    __shared__ int nid_s[TILE_M];

    const int t  = threadIdx.x;
    const int b0 = blockIdx.x * TILE_M;

    if (t < TILE_M) nid_s[t] = node_ids[b0 + t];

    // ---- Phase 1: async DMA tile fill (global -> LDS, no VGPR staging) ----
    // messages tile: 32 rows x 256 f32 = 2048 float4 -> 8 async b128 per thread
#pragma unroll
    for (int i = 0; i < 8; ++i) {
        int e4   = t + i * 256;          // 0..2047
        int row  = e4 >> 6;              // 64 float4 per row
        int col4 = e4 & 63;
        async_load_b128(lds_off(&xs[row * XP + col4 * 4]),
                        messages + (long)(b0 + row) * IN_DIM + col4 * 4);
    }
    // gathered h tile: 32 rows x 128 f32 = 1024 float4 -> 4 async b128 per thread
#pragma unroll
    for (int i = 0; i < 4; ++i) {
        int e4   = t + i * 256;          // 0..1023
        int row  = e4 >> 5;              // 32 float4 per row
        int col4 = e4 & 31;
        int nid  = node_ids[b0 + row];
        async_load_b128(lds_off(&hs[row * HP + col4 * 4]),
                        memory + (long)nid * MEM_DIM + col4 * 4);
    }
    wait_asynccnt0();     // my async DMAs have landed in LDS
    __syncthreads();      // everyone else's too

    // ---- Phase 2: GEMMs via V_WMMA_F32_16X16X4_F32 ----
    const int wave = t >> 5;
    const int lane = t & 31;
    const int lo   = lane & 15;          // M (for A) / N (for B,D) within tile
    const int hi   = lane >> 4;          // K-half selector per ISA layout

    // 48 tile jobs (24 GI, K=256 ++ 24 GH, K=128), 6 per wave.
    // Each job computes BOTH 16-row batch subtiles with dual accumulators, so
    // one weight B-fragment load feeds two WMMAs.
#pragma unroll
    for (int j = 0; j < 6; ++j) {
        int  job  = wave + 8 * j;               // round-robin: 3 GI + 3 GH per wave
        bool isGI = (job < 24);
        int  nt   = isGI ? job : job - 24;      // N-tile index (0..23)
        int  n    = nt * 16 + lo;               // output column 0..383

        const float* W     = isGI ? W_ih : W_hh;
        const float* bias  = isGI ? b_ih : b_hh;
        const float* Abase = isGI ? xs   : hs;
        const int    ap    = isGI ? XP   : HP;
        const int    ldk   = isGI ? IN_DIM : MEM_DIM;
        const int    ksteps = ldk >> 2;         // 64 or 32 K-steps

        float bval = bias[n];
        v8f acc0 = { bval, bval, bval, bval, bval, bval, bval, bval };
        v8f acc1 = acc0;

        // A fragments: lane -> x[M][k + 2*hi + {0,1}]  (ds_load_b64, pitch conflict-free)
        // B fragment : lane -> W[n][k + 2*hi + {0,1}]  (global_load_b64, L2-hot, reused x2)
        const float* arow0 = Abase + (     lo) * ap + 2 * hi;   // rows 0..15
        const float* arow1 = Abase + (16 + lo) * ap + 2 * hi;   // rows 16..31
        const float* wrow  = W + (long)n * ldk + 2 * hi;
#pragma unroll 4
        for (int ks = 0; ks < ksteps; ++ks) {
            v2f b  = *(const v2f*)(wrow  + ks * 4);
            v2f a0 = *(const v2f*)(arow0 + ks * 4);
            v2f a1 = *(const v2f*)(arow1 + ks * 4);
            acc0 = __builtin_amdgcn_wmma_f32_16x16x4_f32(
                       false, a0, false, b, (short)0, acc0, false, false);
            acc1 = __builtin_amdgcn_wmma_f32_16x16x4_f32(
                       false, a1, false, b, (short)0, acc1, false, false);
        }

        float* G = isGI ? gis : ghs;
#pragma unroll
        for (int v = 0; v < 8; ++v) {
            int m = v + 8 * hi;                 // D layout: VGPRv -> M = v + 8*(lane/16)
            G[(     m) * GP + nt * 16 + lo] = acc0[v];
            G[(16 + m) * GP + nt * 16 + lo] = acc1[v];
        }
    }
    __syncthreads();

    // ---- Phase 3: gate math + scatter (2 rows x 8 cols per thread) ----
    const int row_lo = t & 15;
    const int cbase  = (t >> 4) * 8;
#pragma unroll
    for (int msub = 0; msub < 2; ++msub) {
        int row = msub * 16 + row_lo;
        int nid = nid_s[row];
        float* orow = out_mem + (long)nid * MEM_DIM;
#pragma unroll
        for (int i = 0; i < 8; ++i) {
            int c = cbase + i;
            float gir = gis[row * GP + c];
            float ghr = ghs[row * GP + c];
            float giz = gis[row * GP + 128 + c];
            float ghz = ghs[row * GP + 128 + c];
            float gin = gis[row * GP + 256 + c];
            float ghn = ghs[row * GP + 256 + c];
            float rg = 1.0f / (1.0f + __expf(-(gir + ghr)));
            float zg = 1.0f / (1.0f + __expf(-(giz + ghz)));
            float ta = gin + rg * ghn;
            float e2 = __expf(2.0f * ta);
            float ng = 1.0f - 2.0f / (e2 + 1.0f);   // tanh, overflow-safe
            float hv = hs[row * HP + c];
            orow[c] = (1.0f - zg) * ng + zg * hv;
        }
    }

    // ---- Phase 4: timestamp scatter ----
    if (t < TILE_M) {
        out_last[nid_s[t]] = timestamps[b0 + t];
    }
}

// ---------------------------------------------------------------------------
extern "C" void kernel_launch(void* const* d_in, const int* in_sizes, int n_in,
                              void* d_out, int out_size, void* d_ws, size_t ws_size,
                              hipStream_t stream) {
    const int*   node_ids    = (const int*)  d_in[0];
    const float* messages    = (const float*)d_in[1];
    const float* timestamps  = (const float*)d_in[2];
    const float* memory      = (const float*)d_in[3];
    const float* last_update = (const float*)d_in[4];
    const float* W_ih        = (const float*)d_in[5];
    const float* W_hh        = (const float*)d_in[6];
    const float* b_ih        = (const float*)d_in[7];
    const float* b_hh        = (const float*)d_in[8];

    float* out_mem  = (float*)d_out;
    float* out_last = out_mem + (size_t)NUM_NODES * MEM_DIM;

    const long n_mem4  = (long)NUM_NODES * MEM_DIM / 4;  // 16,000,000
    const long n_last4 = NUM_NODES / 4;                  // 125,000
    const long tot     = n_mem4 + n_last4;
    int cgrid = (int)((tot + 255) / 256);
    tgn_copy_kernel<<<cgrid, 256, 0, stream>>>(memory, last_update,
                                               out_mem, out_last, n_mem4, n_last4);

    const int BATCH = 200000;
    tgn_gru_kernel<<<BATCH / TILE_M, 256, 0, stream>>>(
        node_ids, messages, timestamps, memory,
        W_ih, W_hh, b_ih, b_hh, out_mem, out_last);
}